// RNAPocketEncoderV3_45973329936785
// MI455X (gfx1250) — compile-verified
//
#include <hip/hip_runtime.h>
#include <stdint.h>

#define NSC 32
#define NVEC 16
#define NTEN 8
#define ROWF 120
#define NACC 24            // 16 vec norms + 8 ten norms
#define BS 256
#define P1B 1024
#define EPS_LN 1e-5f
#define EPS_NORM 1e-6f

#define GAS __attribute__((address_space(1)))
#define LAS __attribute__((address_space(3)))

#if defined(__gfx1250__) && __has_builtin(__builtin_amdgcn_tensor_load_to_lds) && __has_builtin(__builtin_amdgcn_s_wait_tensorcnt)
#define HAVE_TDM 1
#else
#define HAVE_TDM 0
#endif

#if defined(__gfx1250__) && __has_builtin(__builtin_amdgcn_global_load_async_to_lds_b32)
#define HAVE_ASYNC 1
#else
#define HAVE_ASYNC 0
#endif

typedef uint32_t u32x4_t __attribute__((ext_vector_type(4)));
typedef int      i32x4_t __attribute__((ext_vector_type(4)));
typedef int      i32x8_t __attribute__((ext_vector_type(8)));
typedef float    f32x4_t __attribute__((ext_vector_type(4)));  // native clang vector (NT-store capable)

#if HAVE_TDM
__device__ __forceinline__ uint32_t lds_off_u32(void* p) {
  // generic -> LDS addrspace cast; AS3 pointers are 32-bit LDS byte offsets
  return (uint32_t)(uintptr_t)(LAS void*)p;
}
// 1-D TDM copy of `nelem` 4-byte elements: global -> LDS (ISA ch.8 D# layout)
__device__ __forceinline__ void tdm_load_1d(void* lds_dst, const void* gsrc, uint32_t nelem) {
  uint64_t ga = (uint64_t)(uintptr_t)gsrc;
  u32x4_t g0;
  g0.x = 1u;                                                   // count=1, user desc
  g0.y = lds_off_u32(lds_dst);                                 // lds_addr
  g0.z = (uint32_t)ga;                                         // global_addr[31:0]
  g0.w = (uint32_t)((ga >> 32) & 0x01FFFFFFu) | (2u << 30);    // global_addr[56:32] | type=2
  // group1: data_size=2 (4B) @ [17:16]; tensor_dim0 @ [79:48]; tensor_dim1=1 @ [111:80];
  // tile_dim0 @ [127:112]; tile_dim1=1 @ [143:128]; tensor_dim0_stride @ [207:160]
  uint64_t q0 = ((uint64_t)2u << 16) | ((uint64_t)(nelem & 0xFFFFu) << 48);
  uint64_t q1 = ((uint64_t)((nelem >> 16) & 0xFFFFu)) | ((uint64_t)1u << 16)
              | ((uint64_t)(nelem & 0xFFFFu) << 48);
  uint64_t q2 = ((uint64_t)1u) | ((uint64_t)nelem << 32);
  uint64_t q3 = 0;
  i32x8_t g1;
  g1[0] = (int)(uint32_t)q0; g1[1] = (int)(uint32_t)(q0 >> 32);
  g1[2] = (int)(uint32_t)q1; g1[3] = (int)(uint32_t)(q1 >> 32);
  g1[4] = (int)(uint32_t)q2; g1[5] = (int)(uint32_t)(q2 >> 32);
  g1[6] = (int)(uint32_t)q3; g1[7] = (int)(uint32_t)(q3 >> 32);
  i32x4_t z4 = {0, 0, 0, 0};
#if __clang_major__ >= 23
  i32x8_t z8 = {0, 0, 0, 0, 0, 0, 0, 0};
  __builtin_amdgcn_tensor_load_to_lds(g0, g1, z4, z4, z8, 0);
#else
  __builtin_amdgcn_tensor_load_to_lds(g0, g1, z4, z4, 0);
#endif
}
#endif

// ---------------- Pass 1: per-row norms -> per-block partial sums ----------------
__global__ __launch_bounds__(BS) void pass1_kernel(const float* __restrict__ x,
                                                   float* __restrict__ partial,
                                                   int rows, int nblocks) {
  float acc[NACC];
#pragma unroll
  for (int i = 0; i < NACC; ++i) acc[i] = 0.f;

  const int stride = nblocks * BS;
  for (int r = blockIdx.x * BS + threadIdx.x; r < rows; r += stride) {
    if (r + stride < rows)
      __builtin_prefetch(x + (size_t)(r + stride) * ROWF + NSC, 0, 1);  // global_prefetch_b8
    const f32x4_t* __restrict__ p = (const f32x4_t*)(x + (size_t)r * ROWF + NSC);  // 88 floats, 16B-aligned
    float v[88];
#pragma unroll
    for (int k = 0; k < 22; ++k) {
      f32x4_t q = p[k];
      v[4*k+0] = q.x; v[4*k+1] = q.y; v[4*k+2] = q.z; v[4*k+3] = q.w;
    }
#pragma unroll
    for (int i = 0; i < NVEC; ++i) {
      float a = v[3*i], b = v[3*i+1], c = v[3*i+2];
      acc[i] += fmaxf(sqrtf(a*a + b*b + c*c), EPS_NORM);
    }
#pragma unroll
    for (int i = 0; i < NTEN; ++i) {
      float n2 = 0.f;
#pragma unroll
      for (int j = 0; j < 5; ++j) { float t = v[3*NVEC + 5*i + j]; n2 += t * t; }
      acc[NVEC + i] += fmaxf(sqrtf(n2), EPS_NORM);
    }
  }

  __shared__ float red[BS][NACC + 1];  // +1 pad: conflict-free tree reduce
#pragma unroll
  for (int i = 0; i < NACC; ++i) red[threadIdx.x][i] = acc[i];
  for (int s = BS / 2; s > 0; s >>= 1) {
    __syncthreads();
    if (threadIdx.x < (unsigned)s) {
#pragma unroll
      for (int i = 0; i < NACC; ++i) red[threadIdx.x][i] += red[threadIdx.x + s][i];
    }
  }
  __syncthreads();
  if (threadIdx.x < NACC)
    partial[(size_t)blockIdx.x * NACC + threadIdx.x] = red[0][threadIdx.x];
}

// ---------------- Deterministic final reduction: partials -> 24 means ----------------
__global__ __launch_bounds__(BS) void reduce_kernel(const float* __restrict__ partial,
                                                    float* __restrict__ means,
                                                    int nblocks, float invRows) {
  __shared__ float red[BS];
  for (int a = 0; a < NACC; ++a) {
    float s = 0.f;
    for (int i = threadIdx.x; i < nblocks; i += BS) s += partial[(size_t)i * NACC + a];
    red[threadIdx.x] = s;
    __syncthreads();
    for (int st = BS / 2; st > 0; st >>= 1) {
      if (threadIdx.x < (unsigned)st) red[threadIdx.x] += red[threadIdx.x + st];
      __syncthreads();
    }
    if (threadIdx.x == 0) means[a] = red[0] * invRows;
    __syncthreads();
  }
}

// ---------------- Pass 2: LayerNorm + norm-rescale, streaming rewrite ----------------
// Rows are processed in REVERSE order: pass 1 streamed x front-to-back, so the tail
// of x is MRU in the 192MB L2 when pass 2 starts; reading back-to-front harvests it.
__global__ __launch_bounds__(BS) void pass2_kernel(const float* __restrict__ x,
                                                   const float* __restrict__ w,
                                                   const float* __restrict__ b,
                                                   const float* __restrict__ means,
                                                   float* __restrict__ out, int rows) {
  __shared__ float sw[NSC];
  __shared__ float sb[NSC];
  __shared__ float sm[NACC];
#if HAVE_TDM
  __shared__ float tdm_scratch[1024];  // declared last; generous bound for the TDM tile
#endif

  // Correctness-critical tables via simple, well-defined paths:
  if (threadIdx.x < NSC) { sw[threadIdx.x] = w[threadIdx.x]; sb[threadIdx.x] = b[threadIdx.x]; }

#if HAVE_ASYNC
  // Per-lane async global->LDS copy (one dword per lane, memory -> LDS), ASYNCcnt-tracked.
  if (threadIdx.x < NACC)
    __builtin_amdgcn_global_load_async_to_lds_b32((GAS int*)(means + threadIdx.x),
                                                  (LAS int*)(sm + threadIdx.x), 0, 0);
  if (threadIdx.x < 32) {
#if __has_builtin(__builtin_amdgcn_s_wait_asynccnt)
    __builtin_amdgcn_s_wait_asynccnt(0);
#else
    asm volatile("s_wait_asynccnt 0" ::: "memory");
#endif
  }
#else
  if (threadIdx.x < NACC) sm[threadIdx.x] = means[threadIdx.x];
#endif

#if HAVE_TDM
  // Tensor Data Mover exercise/prefetch of the same 128B table into scratch LDS
  // (wave 0 issues; TDM ignores EXEC). Math does not depend on this copy.
  if (threadIdx.x < 32) {
    tdm_load_1d(tdm_scratch, w, NSC);
    __builtin_amdgcn_s_wait_tensorcnt(0);
  }
#endif
  __syncthreads();

  int idx = blockIdx.x * BS + threadIdx.x;
  if (idx >= rows) return;
  int r = rows - 1 - idx;  // reverse traversal for L2 reuse

  const f32x4_t* __restrict__ p = (const f32x4_t*)(x + (size_t)r * ROWF);
  float v[ROWF];
#pragma unroll
  for (int k = 0; k < 30; ++k) {
    f32x4_t q = p[k];
    v[4*k+0] = q.x; v[4*k+1] = q.y; v[4*k+2] = q.z; v[4*k+3] = q.w;
  }

  // LayerNorm over first 32 features
  float mu = 0.f;
#pragma unroll
  for (int k = 0; k < NSC; ++k) mu += v[k];
  mu *= (1.f / NSC);
  float var = 0.f;
#pragma unroll
  for (int k = 0; k < NSC; ++k) { float d = v[k] - mu; var += d * d; }
  var *= (1.f / NSC);
  float inv = 1.f / sqrtf(var + EPS_LN);
#pragma unroll
  for (int k = 0; k < NSC; ++k) v[k] = (v[k] - mu) * inv * sw[k] + sb[k];

  // 16 x 3-vectors: feats / max(|f|,eps) * mean_n
#pragma unroll
  for (int i = 0; i < NVEC; ++i) {
    float a = v[NSC + 3*i], c = v[NSC + 3*i + 1], d = v[NSC + 3*i + 2];
    float n = fmaxf(sqrtf(a*a + c*c + d*d), EPS_NORM);
    float s = sm[i] / n;
    v[NSC + 3*i] = a * s; v[NSC + 3*i + 1] = c * s; v[NSC + 3*i + 2] = d * s;
  }
  // 8 x 5-tensors
#pragma unroll
  for (int i = 0; i < NTEN; ++i) {
    int base = NSC + 3 * NVEC + 5 * i;
    float n2 = 0.f;
#pragma unroll
    for (int j = 0; j < 5; ++j) n2 += v[base + j] * v[base + j];
    float n = fmaxf(sqrtf(n2), EPS_NORM);
    float s = sm[NVEC + i] / n;
#pragma unroll
    for (int j = 0; j < 5; ++j) v[base + j] *= s;
  }

  // Non-temporal stores: out is never re-read; don't let it evict x from L2.
  f32x4_t* __restrict__ o = (f32x4_t*)(out + (size_t)r * ROWF);
#pragma unroll
  for (int k = 0; k < 30; ++k) {
    f32x4_t q;
    q.x = v[4*k+0]; q.y = v[4*k+1]; q.z = v[4*k+2]; q.w = v[4*k+3];
    __builtin_nontemporal_store(q, o + k);
  }
}

extern "C" void kernel_launch(void* const* d_in, const int* in_sizes, int n_in,
                              void* d_out, int out_size, void* d_ws, size_t ws_size,
                              hipStream_t stream) {
  const float* x = (const float*)d_in[0];
  const float* w = (const float*)d_in[1];
  const float* b = (const float*)d_in[2];
  float* out = (float*)d_out;
  float* ws = (float*)d_ws;

  int rows = in_sizes[0] / ROWF;
  if (rows <= 0) return;

  int grids = (rows + BS - 1) / BS;
  size_t wsf = ws_size / sizeof(float);
  int maxnb = (wsf > (size_t)(2 * NACC)) ? (int)((wsf - NACC) / NACC) : 1;
  int nb = P1B;
  if (nb > grids) nb = grids;
  if (nb > maxnb) nb = maxnb;
  if (nb < 1) nb = 1;

  float* partial = ws;                    // nb * 24 floats, every slot written by pass1
  float* means = ws + (size_t)nb * NACC;  // 24 floats

  pass1_kernel<<<nb, BS, 0, stream>>>(x, partial, rows, nb);
  reduce_kernel<<<1, BS, 0, stream>>>(partial, means, nb, 1.0f / (float)rows);
  pass2_kernel<<<grids, BS, 0, stream>>>(x, w, b, means, out, rows);
}